// JKNet_12412455486110
// MI455X (gfx1250) — compile-verified
//
#include <hip/hip_runtime.h>
#include <math.h>

typedef __attribute__((ext_vector_type(2))) float v2f;
typedef __attribute__((ext_vector_type(8))) float v8f;

#define DIM     256
#define LAYERS  6
#define CLS     40
#define BN_EPS  1e-5f

// ---------------- degree / normalization ----------------

__global__ void k_deg_init(float* __restrict__ deg, int n) {
    int i = blockIdx.x * blockDim.x + threadIdx.x;
    if (i < n) deg[i] = 1.0f;               // self-loop contributes 1
}

__global__ void k_deg_count(const int* __restrict__ dst, float* __restrict__ deg, int e) {
    int i = blockIdx.x * blockDim.x + threadIdx.x;
    if (i < e) atomicAdd(&deg[dst[i]], 1.0f);
}

__global__ void k_dinv(float* __restrict__ deg, int n) {
    int i = blockIdx.x * blockDim.x + threadIdx.x;
    if (i < n) deg[i] = rsqrtf(deg[i]);     // deg >= 1 always
}

// ---------------- WMMA fp32 GEMM: C[M,ncols] = A[M,K] @ B[K,ncols] (+bias) ----
// One wave computes one 16x16 output tile with v_wmma_f32_16x16x4_f32.
// A-layout (ISA 7.12.2, 32-bit A 16x4): lane<16 holds K={0,1} of row M=lane,
// lane>=16 holds K={2,3}. C/D: 8 VGPRs, N=lane&15, M = vgpr + 8*(lane>>4).
// LDB is compile-time so B address math strength-reduces to immediate offsets.
// Out-of-range output columns are CLAMPED (loads stay in-bounds, results for
// those lanes are garbage and simply never stored) -> no EXEC manipulation in
// the hot loop.

template<int LDB>
__global__ __launch_bounds__(256)
void k_gemm_wmma(const float* __restrict__ A, int lda,
                 const float* __restrict__ B,
                 float* __restrict__ C, int ldc,
                 int K, int tiles_n, int ncols,
                 const float* __restrict__ bias, int total_waves)
{
    int gw = blockIdx.x * (blockDim.x >> 5) + (threadIdx.x >> 5);
    if (gw >= total_waves) return;          // uniform per wave: EXEC stays all-1s
    int tm = gw / tiles_n;
    int tn = gw - tm * tiles_n;

    int lane = threadIdx.x & 31;
    int half = lane >> 4;                   // 0: K pair {0,1}, 1: K pair {2,3}
    int l16  = lane & 15;

    const float* Ap = A + (size_t)(tm * 16 + l16) * lda + 2 * half;
    int  bcol = tn * 16 + l16;
    bool bok  = bcol < ncols;
    int  bcol_eff = bok ? bcol : (ncols - 1);      // clamp: keep loads in-bounds
    const float* Bp = B + bcol_eff + (size_t)(2 * half) * LDB;

    v8f acc = {};
#pragma unroll 8
    for (int k = 0; k < K; k += 4) {
        v2f a = *(const v2f*)Ap;            // 8B-aligned: even offset, even lda
        Ap += 4;
        v2f b;
        b.x = Bp[0];
        b.y = Bp[LDB];
        Bp += 4 * LDB;
        acc = __builtin_amdgcn_wmma_f32_16x16x4_f32(
                  false, a, false, b, (short)0, acc, false, false);
    }

    if (bok) {
        float bb = bias ? bias[bcol] : 0.0f;
        int r0 = tm * 16 + half * 8;
#pragma unroll
        for (int v = 0; v < 8; ++v)
            C[(size_t)(r0 + v) * ldc + bcol] = acc[v] + bb;
    }
}

// ---------------- agg init: self-loop term + layer bias ----------------

__global__ void k_prep(const float* __restrict__ g, const float* __restrict__ dinv,
                       const float* __restrict__ bvec, float* __restrict__ agg, int n)
{
    int idx  = blockIdx.x * blockDim.x + threadIdx.x;
    int node = idx >> 6;
    int c4   = (idx & 63) << 2;
    if (node >= n) return;
    float w = dinv[node]; w *= w;
    float4 v = *(const float4*)(g + (size_t)node * DIM + c4);
    float4 b = *(const float4*)(bvec + c4);
    float4 o;
    o.x = w * v.x + b.x; o.y = w * v.y + b.y;
    o.z = w * v.z + b.z; o.w = w * v.w + b.w;
    *(float4*)(agg + (size_t)node * DIM + c4) = o;
}

// ---------------- edge scatter: agg[dst] += norm * g[src] ----------------

__global__ void k_scatter(const int* __restrict__ src, const int* __restrict__ dst,
                          const float* __restrict__ dinv, const float* __restrict__ g,
                          float* __restrict__ agg, int e)
{
    int idx = blockIdx.x * blockDim.x + threadIdx.x;
    int ed  = idx >> 6;
    int c4  = (idx & 63) << 2;
    if (ed >= e) return;
    int s = src[ed], d = dst[ed];
    float w = dinv[s] * dinv[d];
    float4 v = *(const float4*)(g + (size_t)s * DIM + c4);   // L2-resident gather
    float* o = agg + (size_t)d * DIM + c4;
    atomicAdd(o + 0, w * v.x);
    atomicAdd(o + 1, w * v.y);
    atomicAdd(o + 2, w * v.z);
    atomicAdd(o + 3, w * v.w);
}

// ---------------- BatchNorm stats ----------------

__global__ void k_zero_stats(float* __restrict__ sums) {
    int i = blockIdx.x * blockDim.x + threadIdx.x;
    if (i < 2 * DIM) sums[i] = 0.0f;
}

// thread = column (coalesced row reads), block = 128-row chunk, atomic partials
__global__ __launch_bounds__(256)
void k_stats(const float* __restrict__ agg, float* __restrict__ sums, int n)
{
    int c  = threadIdx.x;
    int r0 = blockIdx.x * 128;
    int r1 = r0 + 128; if (r1 > n) r1 = n;
    float s = 0.0f, s2 = 0.0f;
    for (int r = r0; r < r1; ++r) {
        float v = agg[(size_t)r * DIM + c];
        s += v; s2 += v * v;
    }
    atomicAdd(&sums[c], s);
    atomicAdd(&sums[DIM + c], s2);
}

__global__ void k_finalize(const float* __restrict__ sums, const float* __restrict__ gamma,
                           const float* __restrict__ beta, float* __restrict__ ss, float invN)
{
    int c = threadIdx.x;                    // one block of 256
    float mu   = sums[c] * invN;
    float var  = sums[DIM + c] * invN - mu * mu;
    float rstd = rsqrtf(var + BN_EPS);
    float sc   = rstd * gamma[c];
    ss[c]       = sc;                       // scale
    ss[DIM + c] = beta[c] - mu * sc;        // shift
}

// ---------------- fused BN + ReLU -> rep slice ----------------

__global__ void k_apply(const float* __restrict__ agg, const float* __restrict__ ss,
                        float* __restrict__ rep, int n, int layer)
{
    int idx  = blockIdx.x * blockDim.x + threadIdx.x;
    int node = idx >> 6;
    int c4   = (idx & 63) << 2;
    if (node >= n) return;
    float4 a  = *(const float4*)(agg + (size_t)node * DIM + c4);
    float4 sc = *(const float4*)(ss + c4);
    float4 sh = *(const float4*)(ss + DIM + c4);
    float4 o;
    o.x = fmaxf(a.x * sc.x + sh.x, 0.0f);
    o.y = fmaxf(a.y * sc.y + sh.y, 0.0f);
    o.z = fmaxf(a.z * sc.z + sh.z, 0.0f);
    o.w = fmaxf(a.w * sc.w + sh.w, 0.0f);
    *(float4*)(rep + (size_t)node * (LAYERS * DIM) + layer * DIM + c4) = o;
}

// ---------------- driver ----------------

extern "C" void kernel_launch(void* const* d_in, const int* in_sizes, int n_in,
                              void* d_out, int out_size, void* d_ws, size_t ws_size,
                              hipStream_t stream)
{
    const float* x      = (const float*)d_in[0];
    const int*   ei     = (const int*)  d_in[1];
    const float* Ws     = (const float*)d_in[2];
    const float* bs     = (const float*)d_in[3];
    const float* gammas = (const float*)d_in[4];
    const float* betas  = (const float*)d_in[5];
    const float* Wc     = (const float*)d_in[6];
    const float* bc     = (const float*)d_in[7];
    float*       out    = (float*)d_out;

    const int N = in_sizes[0] / DIM;
    const int E = in_sizes[1] / 2;
    const int* srcI = ei;
    const int* dstI = ei + E;

    float* ws = (float*)d_ws;
    size_t o = 0;
    float* dinv = ws + o;  o += (size_t)((N + 3) & ~3);
    float* g    = ws + o;  o += (size_t)N * DIM;
    float* agg  = ws + o;  o += (size_t)N * DIM;
    float* rep  = ws + o;  o += (size_t)N * DIM * LAYERS;
    float* sums = ws + o;  o += 2 * DIM;
    float* ss   = ws + o;  o += 2 * DIM;

    // --- symmetric GCN normalization, recomputed every call ---
    k_deg_init <<<(N + 255) / 256, 256, 0, stream>>>(dinv, N);
    k_deg_count<<<(E + 255) / 256, 256, 0, stream>>>(dstI, dinv, E);
    k_dinv     <<<(N + 255) / 256, 256, 0, stream>>>(dinv, N);

    const int tilesM = (N + 15) / 16;       // N = 50000 -> 3125 exact

    for (int l = 0; l < LAYERS; ++l) {
        const float* A   = (l == 0) ? x : (rep + (size_t)(l - 1) * DIM);
        int          lda = (l == 0) ? DIM : (LAYERS * DIM);

        int tilesN = DIM / 16;
        int waves  = tilesM * tilesN;
        k_gemm_wmma<DIM><<<(waves + 7) / 8, 256, 0, stream>>>(
            A, lda, Ws + (size_t)l * DIM * DIM,
            g, DIM, DIM, tilesN, DIM, nullptr, waves);

        k_prep      <<<(N * 64 + 255) / 256, 256, 0, stream>>>(g, dinv, bs + l * DIM, agg, N);
        k_zero_stats<<<2, 256, 0, stream>>>(sums);
        k_scatter   <<<(E * 64 + 255) / 256, 256, 0, stream>>>(srcI, dstI, dinv, g, agg, E);
        k_stats     <<<(N + 127) / 128, 256, 0, stream>>>(agg, sums, N);
        k_finalize  <<<1, 256, 0, stream>>>(sums, gammas + l * DIM, betas + l * DIM,
                                            ss, 1.0f / (float)N);
        k_apply     <<<(N * 64 + 255) / 256, 256, 0, stream>>>(agg, ss, rep, N, l);
    }

    // --- jumping-knowledge classifier: rep[N,1536] @ Wc[1536,40] + bc ---
    int tilesN = (CLS + 15) / 16;           // 3 tiles, last one column-masked
    int waves  = tilesM * tilesN;
    k_gemm_wmma<CLS><<<(waves + 7) / 8, 256, 0, stream>>>(
        rep, LAYERS * DIM, Wc,
        out, CLS, LAYERS * DIM, tilesN, CLS, bc, waves);
}